// OSDecoder_24077586662035
// MI455X (gfx1250) — compile-verified
//
#include <hip/hip_runtime.h>

typedef __attribute__((ext_vector_type(2))) float v2f;
typedef __attribute__((ext_vector_type(8))) float v8f;

#define NBITS 128
#define KROWS 64
#define LLRMAX 100.0f

__device__ __forceinline__ unsigned int mono_f32(float f) {
    unsigned int u = __float_as_uint(f);
    return (u & 0x80000000u) ? ~u : (u | 0x80000000u);
}

__device__ __forceinline__ unsigned int getbit(const unsigned int* row, int c) {
    return (row[c >> 5] >> (c & 31)) & 1u;
}

__global__ __launch_bounds__(128) void osd2_kernel(const float* __restrict__ llr_g,
                                                   const float* __restrict__ gm_g,
                                                   float* __restrict__ out_g) {
    const int b   = blockIdx.x;
    const int tid = threadIdx.x;

    __shared__ float llr_s[NBITS];      // clipped llr, original order
    __shared__ float llr_sort[NBITS];   // llr after full column permutation
    __shared__ float m_s[NBITS];        // sigma * llr_sort
    __shared__ int   idx_sort[NBITS];   // composed permutation
    __shared__ int   idx_mrb[NBITS];
    __shared__ int   idxArr[NBITS];
    __shared__ int   inv_s[NBITS];
    __shared__ int   piv_s[KROWS];
    __shared__ unsigned long long colbits[NBITS];
    __shared__ unsigned int rowA[KROWS][4];   // working matrix for elimination
    __shared__ unsigned int rowB[KROWS][4];   // gm_mrb (post column permutation)
    __shared__ unsigned int cbits[4];         // base codeword bits
    __shared__ unsigned int wbits[4];         // winner codeword bits
    __shared__ unsigned long long bestkey;

    // ---- load + clip llr; init shared state ----
    float x = llr_g[b * NBITS + tid];
    x = fminf(fmaxf(x, -LLRMAX), LLRMAX);
    llr_s[tid] = x;
    if (tid == 0) bestkey = ~0ull;
    if (tid < 4) cbits[tid] = 0u;
    __syncthreads();

    // ---- stable rank by descending |llr| (== jnp.argsort(-|llr|), stable) ----
    {
        float a = fabsf(x);
        int rank = 0;
        for (int j = 0; j < NBITS; ++j) {
            float aj = fabsf(llr_s[j]);
            rank += (aj > a) || (aj == a && j < tid);
        }
        idx_sort[rank] = tid;
    }
    __syncthreads();

    // ---- permuted column bitmasks of gm ----
    {
        int src = idx_sort[tid];
        unsigned long long cb = 0ull;
        for (int r = 0; r < KROWS; ++r)
            cb |= ((unsigned long long)(gm_g[r * NBITS + src] != 0.0f)) << r;
        colbits[tid] = cb;
    }
    __syncthreads();

    // ---- transpose to row bitsets (64 rows x 128 bits) ----
    {
        int r  = tid & 63;
        int w0 = (tid >> 6) * 2;
        for (int w = w0; w < w0 + 2; ++w) {
            unsigned int acc = 0u;
            for (int bs = 0; bs < 32; ++bs)
                acc |= (unsigned int)((colbits[w * 32 + bs] >> r) & 1ull) << bs;
            rowA[r][w] = acc;
        }
    }
    __syncthreads();

    // ---- GF(2) Gaussian elimination, 64 data-dependent pivot steps ----
    for (int ic = 0; ic < KROWS; ++ic) {
        unsigned int r0 = rowA[ic][0], r1 = rowA[ic][1], r2 = rowA[ic][2], r3 = rowA[ic][3];
        int p;
        if      (r0) p =       __builtin_ctz(r0);
        else if (r1) p = 32  + __builtin_ctz(r1);
        else if (r2) p = 64  + __builtin_ctz(r2);
        else if (r3) p = 96  + __builtin_ctz(r3);
        else         p = 0;                       // argmax of all-zero row -> 0
        if (tid == 0) piv_s[ic] = p;
        if (tid < KROWS && tid != ic) {
            if ((rowA[tid][p >> 5] >> (p & 31)) & 1u) {
                rowA[tid][0] ^= r0; rowA[tid][1] ^= r1;
                rowA[tid][2] ^= r2; rowA[tid][3] ^= r3;
            }
        }
        __syncthreads();
    }

    // ---- build idx_mrb = concat(pivots, stable-argsort(idx)[:64]) ----
    idxArr[tid] = tid;
    __syncthreads();
    if (tid == 0) {
        for (int ic = 0; ic < KROWS; ++ic) idxArr[piv_s[ic]] += NBITS;  // scatter-add (dup pivots accumulate)
    }
    __syncthreads();
    {
        int v = idxArr[tid];
        int rk = 0;
        for (int j = 0; j < NBITS; ++j) {
            int vj = idxArr[j];
            rk += (vj < v) || (vj == v && j < tid);
        }
        if (rk < KROWS) idx_mrb[KROWS + rk] = tid;   // parity columns
    }
    if (tid < KROWS) idx_mrb[tid] = piv_s[tid];
    __syncthreads();

    // ---- permute columns of eliminated matrix -> rowB (gm_mrb) ----
    {
        int r  = tid & 63;
        int w0 = (tid >> 6) * 2;
        for (int w = w0; w < w0 + 2; ++w) {
            unsigned int acc = 0u;
            for (int bs = 0; bs < 32; ++bs) {
                int sc = idx_mrb[w * 32 + bs];
                acc |= getbit(rowA[r], sc) << bs;
            }
            rowB[r][w] = acc;
        }
    }
    // compose permutation: idx_sort_new[c] = idx_sort[idx_mrb[c]]
    int sn = idx_sort[idx_mrb[tid]];
    __syncthreads();
    idx_sort[tid] = sn;
    llr_sort[tid] = llr_s[sn];
    __syncthreads();

    // ---- base codeword c = XOR of rows with hard-decided info bit = 1 ----
    if (tid < KROWS && llr_sort[tid] < 0.0f) {
        for (int w = 0; w < 4; ++w) atomicXor(&cbits[w], rowB[tid][w]);
    }
    __syncthreads();
    // m = sigma * llr_sort  (sigma = 1 - 2*c_bit)
    {
        unsigned int cb = (cbits[tid >> 5] >> (tid & 31)) & 1u;
        m_s[tid] = cb ? -llr_sort[tid] : llr_sort[tid];
    }
    __syncthreads();

    // ---- per-thread local best key, flushed once with a single ds_min_u64 ----
    unsigned long long localBest = ~0ull;

    // t=0 (thread 0) and t=1 (threads 0..63) scores; key = mono(score)<<16 | ordinal
    if (tid == 0) {
        float s = 0.0f;
        for (int c = 0; c < NBITS; ++c) s += m_s[c];
        unsigned long long k0 = ((unsigned long long)mono_f32(s) << 16);
        localBest = (k0 < localBest) ? k0 : localBest;
    }
    if (tid < KROWS) {
        float s = 0.0f;
        for (int c = 0; c < NBITS; ++c) {
            float mv = m_s[c];
            s += getbit(rowB[tid], c) ? -mv : mv;
        }
        unsigned long long k1 =
            ((unsigned long long)mono_f32(s) << 16) | (unsigned long long)(1 + tid);
        localBest = (k1 < localBest) ? k1 : localBest;
    }

    // ---- t=2 scores: Q = (R*diag(m)) * R^T via V_WMMA_F32_16X16X4_F32 ----
    // wave w owns the 16-row strip ti = w; loops tj = 0..3, K in steps of 4.
    {
        const int lane = tid & 31;
        const int ti   = tid >> 5;
        const int iRow = 16 * ti + (lane & 15);
        const int koff = (lane < 16) ? 0 : 2;
        const int mAdd = (lane >= 16) ? 8 : 0;
        for (int tj = 0; tj < 4; ++tj) {
            const int jCol = 16 * tj + (lane & 15);
            v8f acc = {0.f, 0.f, 0.f, 0.f, 0.f, 0.f, 0.f, 0.f};
            for (int kk = 0; kk < 32; ++kk) {
                int k0 = kk * 4 + koff;
                float m0 = m_s[k0], m1 = m_s[k0 + 1];
                v2f A, B;
                A.x = getbit(rowB[iRow], k0)     ? -m0 : m0;   // A(i,k) = rho_i,k * m_k
                A.y = getbit(rowB[iRow], k0 + 1) ? -m1 : m1;
                B.x = getbit(rowB[jCol], k0)     ? -1.0f : 1.0f;   // B(k,j) = rho_j,k
                B.y = getbit(rowB[jCol], k0 + 1) ? -1.0f : 1.0f;
                acc = __builtin_amdgcn_wmma_f32_16x16x4_f32(
                    false, A, false, B, (short)0, acc, false, false);
            }
#pragma unroll
            for (int vr = 0; vr < 8; ++vr) {
                int i = 16 * ti + vr + mAdd;
                int j = 16 * tj + (lane & 15);
                if (i < j) {
                    int po = i * (127 - i) / 2 + (j - i - 1);   // lex ordinal of (i,j)
                    unsigned long long k2 =
                        ((unsigned long long)mono_f32(acc[vr]) << 16) |
                        (unsigned long long)(65 + po);
                    localBest = (k2 < localBest) ? k2 : localBest;
                }
            }
        }
    }
    // single serialized LDS atomic per thread (128 instead of ~2080)
    atomicMin(&bestkey, localBest);
    __syncthreads();

    // ---- decode winner, reconstruct codeword bits exactly via XOR ----
    if (tid == 0) {
        unsigned int w0 = cbits[0], w1 = cbits[1], w2 = cbits[2], w3 = cbits[3];
        int ord = (int)(bestkey & 0xFFFFull);
        if (ord >= 1 && ord <= 64) {
            int i = ord - 1;
            w0 ^= rowB[i][0]; w1 ^= rowB[i][1]; w2 ^= rowB[i][2]; w3 ^= rowB[i][3];
        } else if (ord >= 65) {
            int po = ord - 65;
            int i = 0;
            while (po >= 63 - i) { po -= 63 - i; ++i; }
            int j = i + 1 + po;
            w0 ^= rowB[i][0] ^ rowB[j][0]; w1 ^= rowB[i][1] ^ rowB[j][1];
            w2 ^= rowB[i][2] ^ rowB[j][2]; w3 ^= rowB[i][3] ^ rowB[j][3];
        }
        wbits[0] = w0; wbits[1] = w1; wbits[2] = w2; wbits[3] = w3;
    }
    // ---- inverse permutation = stable argsort of composed idx_sort ----
    {
        int v = idx_sort[tid];
        int rk = 0;
        for (int j = 0; j < NBITS; ++j) {
            int vj = idx_sort[j];
            rk += (vj < v) || (vj == v && j < tid);
        }
        inv_s[rk] = tid;
    }
    __syncthreads();
    {
        int src = inv_s[tid];
        out_g[b * NBITS + tid] = (float)((wbits[src >> 5] >> (src & 31)) & 1u);
    }
}

extern "C" void kernel_launch(void* const* d_in, const int* in_sizes, int n_in,
                              void* d_out, int out_size, void* d_ws, size_t ws_size,
                              hipStream_t stream) {
    const float* llr = (const float*)d_in[0];   // [128, 128] f32
    const float* gm  = (const float*)d_in[1];   // [64, 128] f32 (0/1)
    float* out = (float*)d_out;                 // [128, 128] f32
    (void)in_sizes; (void)n_in; (void)out_size; (void)d_ws; (void)ws_size;
    osd2_kernel<<<dim3(128), dim3(128), 0, stream>>>(llr, gm, out);
}